// BayerNet_31817117729006
// MI455X (gfx1250) — compile-verified
//
#include <hip/hip_runtime.h>
#include <stdint.h>

// Bayer bilinear demosaic, fused single pass.
// Memory-bound: >=512MiB HBM traffic, ~0.4 GFLOP -> async B128 global->LDS
// staging (ASYNCcnt path) + wide NT b128 stores. WMMA intentionally unused
// (AI ~0.8 FLOP/B, far below the matrix-op ridge).

#define IMG_H 4096
#define IMG_W 4096
#define TILE_H 64
#define TILE_W 128
#define LDS_W  136            // interior at cols 4..131 (16B-aligned), halos at 3 / 132
#define LDS_H  (TILE_H + 2)   // 66
#define COL0   3              // LDS col of global col (tx-1); window base = COL0 + 4*pcx
#define NTHREADS 256

#define AS1 __attribute__((address_space(1)))
#define AS3 __attribute__((address_space(3)))

typedef float vfloat4 __attribute__((ext_vector_type(4)));
typedef int   vint4   __attribute__((ext_vector_type(4)));

// Async global->LDS copies (ASYNCcnt-tracked, CDNA5 path).
__device__ __forceinline__ void async_g2l_b32(const float* gsrc, const float* ldst) {
#if __has_builtin(__builtin_amdgcn_global_load_async_to_lds_b32)
  __builtin_amdgcn_global_load_async_to_lds_b32(
      (AS1 int*)(uintptr_t)gsrc,
      (AS3 int*)(uint32_t)(uintptr_t)ldst,   // low 32 bits of generic == LDS addr
      0, 0);
#else
  uint32_t lva = (uint32_t)(uintptr_t)ldst;
  asm volatile("global_load_async_to_lds_b32 %0, %1, off"
               :: "v"(lva), "v"(gsrc) : "memory");
#endif
}

__device__ __forceinline__ void async_g2l_b128(const float* gsrc, const float* ldst) {
#if __has_builtin(__builtin_amdgcn_global_load_async_to_lds_b128)
  __builtin_amdgcn_global_load_async_to_lds_b128(
      (AS1 vint4*)(uintptr_t)gsrc,
      (AS3 vint4*)(uint32_t)(uintptr_t)ldst,
      0, 0);
#else
  uint32_t lva = (uint32_t)(uintptr_t)ldst;
  asm volatile("global_load_async_to_lds_b128 %0, %1, off"
               :: "v"(lva), "v"(gsrc) : "memory");
#endif
}

__device__ __forceinline__ void wait_async_zero() {
#if __has_builtin(__builtin_amdgcn_s_wait_asynccnt)
  __builtin_amdgcn_s_wait_asynccnt(0);
#else
  asm volatile("s_wait_asynccnt 0x0" ::: "memory");
#endif
}

__device__ __forceinline__ int reflect_idx(int v, int n) {
  v = (v < 0) ? -v : v;               // -1 -> 1  (np.pad 'reflect')
  v = (v >= n) ? (2 * n - 2 - v) : v; // n  -> n-2
  return v;
}

__global__ __launch_bounds__(NTHREADS)
void bayer_demosaic_kernel(const float* __restrict__ x, float* __restrict__ y) {
  __shared__ float tile[LDS_H * LDS_W];

  const int tx  = blockIdx.x * TILE_W;
  const int ty  = blockIdx.y * TILE_H;
  const int img = blockIdx.z;
  const int tid = threadIdx.x;

  const float* xin = x + (size_t)img * (size_t)IMG_H * IMG_W;

  // ---- Stage halo tile into LDS ----
  // Bulk interior: 66 rows x 32 B128 chunks (cols tx..tx+127 -> LDS cols 4..131).
  for (int idx = tid; idx < LDS_H * 32; idx += NTHREADS) {
    const int row = idx >> 5;
    const int ch  = idx & 31;
    const int gr  = reflect_idx(ty - 1 + row, IMG_H);
    async_g2l_b128(xin + (size_t)gr * IMG_W + tx + 4 * ch,
                   &tile[row * LDS_W + 4 + 4 * ch]);
  }
  // Halo columns: 66 rows x 2 single-dword loads (with column reflect).
  for (int idx = tid; idx < LDS_H * 2; idx += NTHREADS) {
    const int row  = idx >> 1;
    const int side = idx & 1;
    const int gr = reflect_idx(ty - 1 + row, IMG_H);
    const int gc = reflect_idx(side ? (tx + TILE_W) : (tx - 1), IMG_W);
    const int lc = side ? (TILE_W + 4) : COL0;     // 132 or 3
    async_g2l_b32(xin + (size_t)gr * IMG_W + gc, &tile[row * LDS_W + lc]);
  }
  wait_async_zero();
  __syncthreads();

  // ---- Compute: each thread does four 2x4 output patches ----
  float* yout = y + (size_t)img * 3u * (size_t)IMG_H * IMG_W;
  const size_t cstride = (size_t)IMG_H * IMG_W;

  const int pcx = tid & 31;   // patch col 0..31  (patch = 2 rows x 4 cols)
  const int pr0 = tid >> 5;   // 0..7

#pragma unroll
  for (int k = 0; k < 4; ++k) {
    const int pry = pr0 + (k << 3);   // 0..31

    // 4x6 LDS window covering the padded neighborhood of this 2x4 patch
    float p[4][6];
#pragma unroll
    for (int r = 0; r < 4; ++r)
#pragma unroll
      for (int c = 0; c < 6; ++c)
        p[r][c] = tile[(2 * pry + r) * LDS_W + 4 * pcx + COL0 + c];

#pragma unroll
    for (int rr = 0; rr < 2; ++rr) {
      float rv[4], gv[4], bv[4];
#pragma unroll
      for (int cc = 0; cc < 4; ++cc) {
        const int r1 = rr + 1, c1 = cc + 1;
        const float center = p[r1][c1];
        const float plus = 0.25f * (p[rr][c1] + p[rr + 2][c1] + p[r1][cc] + p[r1][cc + 2]);
        const float diag = 0.25f * (p[rr][cc] + p[rr][cc + 2] + p[rr + 2][cc] + p[rr + 2][cc + 2]);
        const float havg = 0.5f * (p[r1][cc] + p[r1][cc + 2]);
        const float vavg = 0.5f * (p[rr][c1] + p[rr + 2][c1]);
        if (rr == 0) {
          if ((cc & 1) == 0) { rv[cc] = diag;   gv[cc] = plus;   bv[cc] = center; }
          else               { rv[cc] = vavg;   gv[cc] = center; bv[cc] = havg;   }
        } else {
          if ((cc & 1) == 0) { rv[cc] = havg;   gv[cc] = center; bv[cc] = vavg;   }
          else               { rv[cc] = center; gv[cc] = plus;   bv[cc] = diag;   }
        }
      }
      const int row = ty + 2 * pry + rr;
      const int col = tx + 4 * pcx;           // multiple of 4 -> 16B aligned
      const size_t obase = (size_t)row * IMG_W + col;
      vfloat4 R4 = {rv[0], rv[1], rv[2], rv[3]};
      vfloat4 G4 = {gv[0], gv[1], gv[2], gv[3]};
      vfloat4 B4 = {bv[0], bv[1], bv[2], bv[3]};
      __builtin_nontemporal_store(R4, (vfloat4*)(yout + obase));
      __builtin_nontemporal_store(G4, (vfloat4*)(yout + cstride + obase));
      __builtin_nontemporal_store(B4, (vfloat4*)(yout + 2 * cstride + obase));
    }
  }
}

extern "C" void kernel_launch(void* const* d_in, const int* in_sizes, int n_in,
                              void* d_out, int out_size, void* d_ws, size_t ws_size,
                              hipStream_t stream) {
  (void)n_in; (void)out_size; (void)d_ws; (void)ws_size;
  const float* x = (const float*)d_in[0];
  float* y = (float*)d_out;
  const int nimg = in_sizes[0] / (IMG_H * IMG_W);   // batch (2)
  dim3 grid(IMG_W / TILE_W, IMG_H / TILE_H, nimg);
  dim3 block(NTHREADS);
  bayer_demosaic_kernel<<<grid, block, 0, stream>>>(x, y);
}